// MyCompositor_42219528520133
// MI455X (gfx1250) — compile-verified
//
#include <hip/hip_runtime.h>
#include <cstddef>

// Pixel-parallel front-to-back alpha compositing (pulsar-style).
// Memory-bound streaming kernel: ~175 MB HBM traffic -> ~7.5 us at 23.3 TB/s.
// No matmul structure exists (per-pixel gathered colors, zero operand reuse),
// so WMMA is inapplicable; we optimize the data-movement path instead:
// B128 nontemporal streams for fragments/alphas/output, EXEC-predicated
// gathers from the L2-resident point-cloud table, global_prefetch_b8 ahead.

typedef __attribute__((ext_vector_type(4))) float floatx4;
typedef __attribute__((ext_vector_type(4))) int   intx4;

constexpr int Bn = 4;
constexpr int Kn = 16;
constexpr int Hn = 480;
constexpr int Wn = 640;
constexpr int Cn = 3;
constexpr int Pn = 200000;
constexpr int HWn = Hn * Wn;                    // 307200 (divisible by 4)
constexpr int QUADS_PER_IMG = HWn / 4;          // 76800
constexpr int TOTAL_QUADS = Bn * QUADS_PER_IMG; // 307200

__global__ __launch_bounds__(256)
void composite_quads(const int*   __restrict__ fragments,  // (B,K,H,W) int32
                     const float* __restrict__ alphas,     // (B,K,H,W)
                     const float* __restrict__ ptclds,     // (C,P)
                     const float* __restrict__ im,         // (C,H,W)
                     float*       __restrict__ out)        // (B,C,H,W)
{
    const int q = blockIdx.x * blockDim.x + threadIdx.x;
    if (q >= TOTAL_QUADS) return;

    const int b = q / QUADS_PER_IMG;
    const int r = (q - b * QUADS_PER_IMG) * 4;   // pixel offset within HxW plane

    const size_t planeBase = (size_t)b * Kn * HWn + r;     // 16B aligned (r % 4 == 0)
    const intx4*   fptr = (const intx4*)(fragments + planeBase);
    const floatx4* aptr = (const floatx4*)(alphas + planeBase);
    constexpr size_t kStride = HWn / 4;          // int4/float4 stride between k-planes

    float T[4]       = {1.0f, 1.0f, 1.0f, 1.0f}; // transmittance Π(1-a_j), j<k
    float acc[Cn][4] = {};
    int   bg[4]      = {};

    #pragma unroll
    for (int k = 0; k < Kn; ++k) {
        if (k + 1 < Kn) {
            // gfx1250 global_prefetch_b8: pull next k-plane's cachelines early
            __builtin_prefetch((const void*)(fptr + (size_t)(k + 1) * kStride), 0, 0);
            __builtin_prefetch((const void*)(aptr + (size_t)(k + 1) * kStride), 0, 0);
        }
        // Read-once streams: nontemporal so they don't evict the ptclds table
        const intx4   f4 = __builtin_nontemporal_load(fptr + (size_t)k * kStride);
        const floatx4 a4 = __builtin_nontemporal_load(aptr + (size_t)k * kStride);

        #pragma unroll
        for (int j = 0; j < 4; ++j) {
            const int   fr = f4[j];
            const float av = (fr >= 0) ? a4[j] : 0.0f;   // a = alpha * valid
            if (k == 0) bg[j] = (fr < 0);
            const float wk = av * T[j];                  // a_k * excl-cumprod
            T[j] = T[j] * (1.0f - av);                   // matches cumprod order
            // Skip gather+FMA when weight is exactly zero (bit-exact: adds +/-0).
            // Per-lane predication -> EXEC-masked, drops ~25% of L2 gather requests.
            if (av > 0.0f) {
                acc[0][j] = fmaf(ptclds[fr],          wk, acc[0][j]);
                acc[1][j] = fmaf(ptclds[Pn + fr],     wk, acc[1][j]);
                acc[2][j] = fmaf(ptclds[2 * Pn + fr], wk, acc[2][j]);
            }
        }
    }

    // Background fallback where fragments[:,0] < 0 (load im only if needed)
    floatx4 imv[Cn] = {};
    if (bg[0] | bg[1] | bg[2] | bg[3]) {
        #pragma unroll
        for (int c = 0; c < Cn; ++c)
            imv[c] = *(const floatx4*)(im + (size_t)c * HWn + r);
    }

    #pragma unroll
    for (int c = 0; c < Cn; ++c) {
        floatx4 o;
        #pragma unroll
        for (int j = 0; j < 4; ++j)
            o[j] = bg[j] ? imv[c][j] : acc[c][j];
        __builtin_nontemporal_store(
            o, (floatx4*)(out + (size_t)(b * Cn + c) * HWn + r));
    }
}

extern "C" void kernel_launch(void* const* d_in, const int* in_sizes, int n_in,
                              void* d_out, int out_size, void* d_ws, size_t ws_size,
                              hipStream_t stream) {
    (void)in_sizes; (void)n_in; (void)out_size; (void)d_ws; (void)ws_size;
    const int*   fragments = (const int*)  d_in[0];
    const float* alphas    = (const float*)d_in[1];
    const float* ptclds    = (const float*)d_in[2];
    const float* im        = (const float*)d_in[3];
    float*       out       = (float*)d_out;

    const int threads = 256;
    const int blocks  = (TOTAL_QUADS + threads - 1) / threads;  // 1200
    composite_quads<<<blocks, threads, 0, stream>>>(fragments, alphas, ptclds, im, out);
}